// Caser_8658654068805
// MI455X (gfx1250) — compile-verified
//
#include <hip/hip_runtime.h>
#include <hip/hip_bf16.h>

// Sizes (fixed by the reference)
#define BB 4096
#define LL 5
#define DD 128
#define NHH 16
#define NVV 4
#define TT 100
#define FC1 752   // NV*D + NH*(5+4+3+2+1)

typedef float v2f __attribute__((ext_vector_type(2)));
typedef float v8f __attribute__((ext_vector_type(8)));

// ---------------------------------------------------------------------------
// Kernel 1: embedding gather + vertical/horizontal convs -> z (B,752)
//           also writes uemb into xc[:,128:256]
// One block of 128 threads per batch row (4 wave32s).
// ---------------------------------------------------------------------------
__global__ void caser_features(const int* __restrict__ seq,
                               const int* __restrict__ user,
                               const float* __restrict__ item_table,
                               const float* __restrict__ user_table,
                               const float* __restrict__ Wv,
                               const float* __restrict__ bv,
                               const float* __restrict__ Wh,
                               const float* __restrict__ bh,
                               float* __restrict__ z,
                               float* __restrict__ xc)
{
    const int b   = blockIdx.x;
    const int tid = threadIdx.x;          // 0..127  == d
    const int wave = tid >> 5;
    const int lane = tid & 31;

    __shared__ float semb[LL][DD];        // 2.5 KB
    __shared__ float sWv[NVV * LL];       // 80 B

    // gather item embeddings for this row; keep in regs for vertical conv
    float e[LL];
    #pragma unroll
    for (int t = 0; t < LL; ++t) {
        const int it = seq[b * LL + t];
        e[t] = item_table[(size_t)it * DD + tid];
        semb[t][tid] = e[t];
    }
    if (tid < NVV * LL) sWv[tid] = Wv[tid];
    __syncthreads();

    // vertical conv: ver[v][d] = sum_t e[t]*Wv[v,t] + bv[v]  -> z[:, v*128+d]
    #pragma unroll
    for (int v = 0; v < NVV; ++v) {
        float acc = bv[v];
        #pragma unroll
        for (int t = 0; t < LL; ++t) acc += e[t] * sWv[v * LL + t];
        z[(size_t)b * FC1 + v * DD + tid] = acc;
    }

    // user embedding -> second half of xc
    xc[(size_t)b * (2 * DD) + DD + tid] =
        user_table[(size_t)user[b] * DD + tid];

    // horizontal convs: 240 outputs total (sum_l NH*(6-l)), one wave per output
    for (int o = wave; o < 240; o += 4) {
        // decode o -> (l, f, t) and z-offset of this l's group
        int rem = o;
        int off = NVV * DD;               // 512
        int l = 1;
        for (; l <= LL; ++l) {
            const int cnt = NHH * (LL - l + 1);
            if (rem < cnt) break;
            rem -= cnt; off += cnt;
        }
        const int lout = LL - l + 1;
        const int f = rem / lout;
        const int t = rem % lout;

        // dot over (s in [0,l)) x (d in [0,128))
        const float* wrow = Wh + ((size_t)(l - 1) * NHH + f) * (LL * DD);
        float acc = 0.f;
        for (int i = lane; i < l * DD; i += 32) {
            const int s = i >> 7;
            const int d = i & (DD - 1);
            acc += semb[t + s][d] * wrow[s * DD + d];
        }
        // wave32 reduction
        #pragma unroll
        for (int sft = 16; sft >= 1; sft >>= 1)
            acc += __shfl_down(acc, sft, 32);
        if (lane == 0) {
            acc += bh[(l - 1) * NHH + f];
            z[(size_t)b * FC1 + off + f * lout + t] = fmaxf(acc, 0.f);
        }
    }
}

// ---------------------------------------------------------------------------
// Kernel 2: x = relu(z @ W1 + b1) via V_WMMA_F32_16X16X4_F32 (fp32 WMMA).
// Block = 256 threads (8 waves). Each block: 16 batch rows; wave w owns the
// 16x16 output tile at columns [16w,16w+16). K loop: 752/4 = 188 WMMAs/wave.
// A (z tile) staged in padded LDS; B (W1, 385 KB) streamed from L2.
// Writes into xc[:, 0:128].
// ---------------------------------------------------------------------------
__global__ void caser_fc1(const float* __restrict__ z,
                          const float* __restrict__ W1,
                          const float* __restrict__ b1,
                          float* __restrict__ xc)
{
    __shared__ float sz[16][FC1 + 4];     // pad 752->756: conflict-free col reads
    const int tid  = threadIdx.x;         // 0..255
    const int wave = tid >> 5;
    const int lane = tid & 31;
    const int m0   = blockIdx.x * 16;

    // cooperative load of the 16x752 A tile
    for (int i = tid; i < 16 * FC1; i += 256) {
        const int r = i / FC1, c = i % FC1;
        sz[r][c] = z[(size_t)(m0 + r) * FC1 + c];
    }
    __syncthreads();

    const int n0   = wave * 16;
    const int mrow = lane & 15;
    const int half = lane >> 4;           // 0: K={0,1}, 1: K={2,3}

    v8f acc = {};
    for (int k0 = 0; k0 < FC1; k0 += 4) {
        const int ka = k0 + half * 2;
        v2f a, bf;
        // A 16x4 fp32 layout: lanes 0-15 M=0..15 K0/K1; lanes 16-31 K2/K3
        a.x = sz[mrow][ka + 0];
        a.y = sz[mrow][ka + 1];
        // B 4x16 fp32 layout: row K striped over lanes (N = lane%16)
        bf.x = W1[(size_t)(ka + 0) * DD + n0 + mrow];
        bf.y = W1[(size_t)(ka + 1) * DD + n0 + mrow];
        acc = __builtin_amdgcn_wmma_f32_16x16x4_f32(
            /*neg_a=*/false, a, /*neg_b=*/false, bf,
            /*c_mod=*/(short)0, acc, /*reuse_a=*/false, /*reuse_b=*/false);
    }

    // D layout: VGPR r -> (M=r, N=lane) lanes 0-15 ; (M=r+8, N=lane-16) 16-31
    const int n = n0 + (lane & 15);
    const float bias = b1[n];
    #pragma unroll
    for (int r = 0; r < 8; ++r) {
        const int m = m0 + r + half * 8;
        const float v = acc[r] + bias;
        xc[(size_t)m * (2 * DD) + n] = fmaxf(v, 0.f);
    }
}

// ---------------------------------------------------------------------------
// Kernel 3: res[b,t] = dot(W2_table[items[b,t]], xc[b]) + b2[items[b,t]]
// One block (256 thr, 8 waves) per batch row; xc row cached in LDS; each wave
// handles t = wave, wave+8, ... with 2x float4 loads/lane + wave32 reduce.
// global_prefetch_b8 hints for the next gathered row.
// ---------------------------------------------------------------------------
__global__ void caser_score(const int* __restrict__ items,
                            const float* __restrict__ xc,
                            const float* __restrict__ W2,
                            const float* __restrict__ b2,
                            float* __restrict__ out)
{
    const int b    = blockIdx.x;
    const int tid  = threadIdx.x;         // 0..255
    const int wave = tid >> 5;
    const int lane = tid & 31;

    __shared__ __align__(16) float sxc[2 * DD];
    sxc[tid] = xc[(size_t)b * (2 * DD) + tid];
    __syncthreads();

    // each lane owns 8 contiguous xc elements
    const float4 xa = *(const float4*)&sxc[lane * 8 + 0];
    const float4 xb = *(const float4*)&sxc[lane * 8 + 4];

    for (int t = wave; t < TT; t += 8) {
        const int it = items[b * TT + t];
        // prefetch the row for the next iteration of this wave
        const int tn = t + 8;
        if (tn < TT) {
            const int itn = items[b * TT + tn];
            __builtin_prefetch(W2 + (size_t)itn * (2 * DD), 0, 1);
        }
        const float4* wrow = (const float4*)(W2 + (size_t)it * (2 * DD));
        const float4 wa = wrow[lane * 2 + 0];
        const float4 wb = wrow[lane * 2 + 1];
        float acc = wa.x * xa.x + wa.y * xa.y + wa.z * xa.z + wa.w * xa.w
                  + wb.x * xb.x + wb.y * xb.y + wb.z * xb.z + wb.w * xb.w;
        #pragma unroll
        for (int s = 16; s >= 1; s >>= 1)
            acc += __shfl_down(acc, s, 32);
        if (lane == 0)
            out[(size_t)b * TT + t] = acc + b2[it];
    }
}

// ---------------------------------------------------------------------------
extern "C" void kernel_launch(void* const* d_in, const int* in_sizes, int n_in,
                              void* d_out, int out_size, void* d_ws, size_t ws_size,
                              hipStream_t stream) {
    const int*   seq        = (const int*)  d_in[0];
    const int*   user       = (const int*)  d_in[1];
    const int*   items      = (const int*)  d_in[2];
    const float* item_table = (const float*)d_in[3];
    const float* user_table = (const float*)d_in[4];
    const float* Wv         = (const float*)d_in[5];
    const float* bv         = (const float*)d_in[6];
    const float* Wh         = (const float*)d_in[7];
    const float* bh         = (const float*)d_in[8];
    const float* W1         = (const float*)d_in[9];
    const float* b1         = (const float*)d_in[10];
    const float* W2         = (const float*)d_in[11];
    const float* b2         = (const float*)d_in[12];
    float* out = (float*)d_out;

    // workspace: z (B x 752) then xc (B x 256)  ~ 16.4 MB
    float* z  = (float*)d_ws;
    float* xc = z + (size_t)BB * FC1;

    caser_features<<<BB, 128, 0, stream>>>(seq, user, item_table, user_table,
                                           Wv, bv, Wh, bh, z, xc);
    caser_fc1<<<BB / 16, 256, 0, stream>>>(z, W1, b1, xc);
    caser_score<<<BB, 256, 0, stream>>>(items, xc, W2, b2, out);
}